// HumanContact3DPredictor_51737176048098
// MI455X (gfx1250) — compile-verified
//
#include <hip/hip_runtime.h>
#include <stdint.h>

// Problem constants (match reference)
#define NV      6890
#define NB      8
#define NVIEW   4
#define NH      512
#define NW      512
#define VHW     (NVIEW * NH * NW)          // 1,048,576 pixels
#define THRESH  0.3f

// Tiling
#define BLOCKS_PER_BATCH 64
#define CHUNK   (VHW / BLOCKS_PER_BATCH)   // 16384 pixels per block
#define TILE    256                        // pixels per tile == blockDim.x
#define NTILES  (CHUNK / TILE)             // 64
#define TILE_DW (TILE * 3)                 // 768 dwords per idx/bary tile

typedef __attribute__((ext_vector_type(4))) unsigned int u32x4;
typedef __attribute__((ext_vector_type(8))) int          i32x8;
typedef __attribute__((ext_vector_type(4))) int          i32x4;

// ---------------------------------------------------------------------------
// TDM: issue a 1-D tensor DMA of `ndw` dwords from global -> LDS.
// D# built per CDNA5 ISA section 8.3/8.4:
//   group0: count=1 | lds_addr | global_addr[56:0] | type=2 ("image")
//   group1: data_size=4B, tensor_dim0 = tile_dim0 = ndw, tensor_dim1 = tile_dim1 = 1
//   groups 2/3 (+ trailing group on this toolchain's 6-arg builtin): zero
// Per-wave instruction (EXEC ignored); tracked with TENSORcnt.
// ---------------------------------------------------------------------------
__device__ __forceinline__ void tdm_load_1d(uint32_t lds_off, const void* gptr,
                                            uint32_t ndw) {
  const uint64_t ga = (uint64_t)(uintptr_t)gptr;
  u32x4 g0;
  g0[0] = 1u;                                               // count=1, user mode
  g0[1] = lds_off;                                          // LDS byte address
  g0[2] = (uint32_t)ga;                                     // global_addr[31:0]
  g0[3] = ((uint32_t)(ga >> 32) & 0x01FFFFFFu) | (2u << 30); // ga[56:32] | type=2

  i32x8 g1;
  g1[0] = (int)(2u << 16);                       // data_size = 2 -> 4 bytes
  g1[1] = (int)((ndw & 0xFFFFu) << 16);          // tensor_dim0[15:0] (abar addr = 0)
  g1[2] = (int)((ndw >> 16) | (1u << 16));       // tensor_dim0[31:16] | tensor_dim1=1
  g1[3] = (int)((ndw & 0xFFFFu) << 16);          // tile_dim0 = ndw
  g1[4] = 1;                                     // tile_dim1 = 1, tile_dim2 = 0
  g1[5] = (int)ndw;                              // tensor_dim0_stride[31:0]
  g1[6] = 0;                                     // stride0[47:32] | stride1[15:0]
  g1[7] = 0;

  i32x4 z4 = {0, 0, 0, 0};
  i32x8 z8 = {0, 0, 0, 0, 0, 0, 0, 0};
  __builtin_amdgcn_tensor_load_to_lds(g0, g1, z4, z4, z8, 0);
}

// LDS byte offset of a generic pointer into shared memory (addrspacecast
// LDS->generic keeps the wave-relative LDS offset in the low 32 bits).
__device__ __forceinline__ uint32_t lds_offset(const void* p) {
  return (uint32_t)(uintptr_t)p;
}

// ---------------------------------------------------------------------------
// Scatter kernel. gridDim = (BLOCKS_PER_BATCH, NB). One batch per block row.
// LDS layout (dynamic):
//   [0,                 NV*4)          s_pred  f32[6890]
//   [NV*4,              2*NV*4)        s_cnt   u32[6890]
//   [2*NV*4,            +2*TILE_DW*4)  s_idx   i32[2][768]  (double buffered)
//   [...,               +2*TILE_DW*4)  s_bary  f32[2][768]  (double buffered)
// idx/bary tiles are streamed in via the Tensor Data Mover, double-buffered
// against the ds-atomic histogram inner loop. The tile-0 DMA is issued FIRST
// so it overlaps with histogram zero-init (staging buffers are disjoint from
// the histograms and only consumed after s_wait_tensorcnt + barrier).
// ---------------------------------------------------------------------------
extern "C" __global__ __launch_bounds__(256)
void hc3d_scatter(const float* __restrict__ seg,
                  const int* __restrict__ p2v,
                  const float* __restrict__ bary,
                  float* __restrict__ g_pred,
                  unsigned int* __restrict__ g_cnt) {
  extern __shared__ unsigned char smem[];
  float*        s_pred = (float*)smem;
  unsigned int* s_cnt  = (unsigned int*)(smem + (size_t)NV * 4);
  int*          s_idx  = (int*)  (smem + (size_t)2 * NV * 4);
  float*        s_bary = (float*)(smem + (size_t)2 * NV * 4 + (size_t)2 * TILE_DW * 4);

  const int tid = threadIdx.x;
  const int b   = blockIdx.y;
  const int p0  = blockIdx.x * CHUNK;

  const bool issuer = (tid < 32);   // wave 0 drives the TDM (uniform per wave)

  // Prologue: DMA tile 0 into buffer 0 (in flight while we zero histograms).
  if (issuer) {
    tdm_load_1d(lds_offset(s_idx),  p2v  + (size_t)p0 * 3, TILE_DW);
    tdm_load_1d(lds_offset(s_bary), bary + (size_t)p0 * 3, TILE_DW);
  }

  // Zero private histograms.
  for (int i = tid; i < NV; i += 256) { s_pred[i] = 0.0f; s_cnt[i] = 0u; }
  __syncthreads();   // histogram zeroing complete (TDM still in flight)

  for (int t = 0; t < NTILES; ++t) {
    const int cur = t & 1;

    if (issuer) {
      if (t + 1 < NTILES) {
        const int    nxt  = (t + 1) & 1;
        const size_t base = (size_t)(p0 + (t + 1) * TILE) * 3;
        tdm_load_1d(lds_offset(s_idx  + nxt * TILE_DW), p2v  + base, TILE_DW);
        tdm_load_1d(lds_offset(s_bary + nxt * TILE_DW), bary + base, TILE_DW);
        // 4 outstanding max; <=2 left  ==>  tile t's two DMAs have completed
        __builtin_amdgcn_s_wait_tensorcnt(2);
      } else {
        __builtin_amdgcn_s_wait_tensorcnt(0);
      }
    }
    __syncthreads();   // tile t visible to all waves

    // One pixel per thread per tile.
    const int   base = cur * TILE_DW + tid * 3;
    const int   i0 = s_idx[base + 0];
    const int   i1 = s_idx[base + 1];
    const int   i2 = s_idx[base + 2];
    const float w0 = s_bary[base + 0];
    const float w1 = s_bary[base + 1];
    const float w2 = s_bary[base + 2];
    const bool valid = ((unsigned)i0 < (unsigned)NV) &&
                       ((unsigned)i1 < (unsigned)NV) &&
                       ((unsigned)i2 < (unsigned)NV);

    const int   p  = p0 + t * TILE + tid;
    const float sv = seg[(size_t)b * VHW + p];

    if (valid && (sv > THRESH)) {
      atomicAdd(&s_pred[i0], w0); atomicAdd(&s_cnt[i0], 1u);
      atomicAdd(&s_pred[i1], w1); atomicAdd(&s_cnt[i1], 1u);
      atomicAdd(&s_pred[i2], w2); atomicAdd(&s_cnt[i2], 1u);
    }
    __syncthreads();   // all waves done with buf[cur] before it is re-filled
  }

  // Flush private histogram to the global (L2-resident) accumulator.
  for (int i = tid; i < NV; i += 256) {
    const unsigned int c = s_cnt[i];
    if (c) {
      atomicAdd(&g_pred[(size_t)b * NV + i], s_pred[i]);
      atomicAdd(&g_cnt [(size_t)b * NV + i], c);
    }
  }
}

// ---------------------------------------------------------------------------
// Zero the global accumulators (pred f32 + cnt u32) in d_ws.
// ---------------------------------------------------------------------------
extern "C" __global__ void hc3d_zero(float* __restrict__ g_pred,
                                     unsigned int* __restrict__ g_cnt, int n) {
  int i = blockIdx.x * blockDim.x + threadIdx.x;
  if (i < n) { g_pred[i] = 0.0f; g_cnt[i] = 0u; }
}

// ---------------------------------------------------------------------------
// Normalize + threshold -> out[8][6890].
// ---------------------------------------------------------------------------
extern "C" __global__ void hc3d_finalize(const float* __restrict__ g_pred,
                                         const unsigned int* __restrict__ g_cnt,
                                         float* __restrict__ out, int n) {
  int i = blockIdx.x * blockDim.x + threadIdx.x;
  if (i < n) {
    float p = g_pred[i];
    const unsigned int c = g_cnt[i];
    if (c > 0u) p = p / (float)c;
    out[i] = (p > THRESH) ? 1.0f : 0.0f;
  }
}

// ---------------------------------------------------------------------------
extern "C" void kernel_launch(void* const* d_in, const int* in_sizes, int n_in,
                              void* d_out, int out_size, void* d_ws, size_t ws_size,
                              hipStream_t stream) {
  (void)in_sizes; (void)n_in; (void)out_size; (void)ws_size;

  const float* seg  = (const float*)d_in[0];   // [8,4,512,512] f32
  const int*   p2v  = (const int*)  d_in[1];   // [4,512,512,3] i32
  const float* bary = (const float*)d_in[2];   // [4,512,512,3] f32
  float*       out  = (float*)d_out;           // [8,6890] f32

  float*        g_pred = (float*)d_ws;                                   // 8*6890 f32
  unsigned int* g_cnt  = (unsigned int*)((char*)d_ws +
                                         (size_t)NB * NV * sizeof(float)); // 8*6890 u32

  const int n_acc = NB * NV;                       // 55,120
  hc3d_zero<<<(n_acc + 255) / 256, 256, 0, stream>>>(g_pred, g_cnt, n_acc);

  const size_t smem = (size_t)2 * NV * 4 + (size_t)4 * TILE_DW * 4;  // 67,408 B
  hc3d_scatter<<<dim3(BLOCKS_PER_BATCH, NB), 256, smem, stream>>>(
      seg, p2v, bary, g_pred, g_cnt);

  hc3d_finalize<<<(n_acc + 255) / 256, 256, 0, stream>>>(g_pred, g_cnt, out, n_acc);
}